// GNN_node_54374285967975
// MI455X (gfx1250) — compile-verified
//
#include <hip/hip_runtime.h>
#include <math.h>

#define NN    10000
#define EE    320000
#define DD    128
#define LL    5
#define NHID  20      // CHID
#define WHID  64
#define BONDV 16
#define TOPK  100

typedef __attribute__((ext_vector_type(2))) float v2f;
typedef __attribute__((ext_vector_type(8))) float v8f;

// ---------------------------------------------------------------- embeddings
__global__ void k_embed(const int* __restrict__ x_feat,
                        const float* __restrict__ atom_emb,
                        float* __restrict__ x, float* __restrict__ nrep)
{
    int i = blockIdx.x * blockDim.x + threadIdx.x;
    if (i >= NN * DD) return;
    int n = i >> 7, d = i & 127;
    float v = atom_emb[x_feat[n * 2 + 0] * DD + d]
            + atom_emb[128 * DD + x_feat[n * 2 + 1] * DD + d];
    x[i] = v;
    nrep[i] = v;           // node_rep initialized with h_list[0] = x
}

// ------------------------------------------------------- tiny node MLPs (wave/node)
__device__ __forceinline__ float mlp_head(const float xr[4], int kb,
                                          const float* __restrict__ w1,
                                          const float* __restrict__ b1,
                                          const float* __restrict__ w2,
                                          const float* __restrict__ b2)
{
    float p[NHID];
#pragma unroll
    for (int c = 0; c < NHID; ++c) p[c] = 0.f;
#pragma unroll
    for (int j = 0; j < 4; ++j) {
        float xv = xr[j];
        const float* wrow = w1 + (kb + j) * NHID;
#pragma unroll
        for (int c = 0; c < NHID; ++c) p[c] += xv * wrow[c];
    }
#pragma unroll
    for (int c = 0; c < NHID; ++c) {
        float v = p[c];
#pragma unroll
        for (int o = 16; o > 0; o >>= 1) v += __shfl_xor(v, o, 32);
        p[c] = v;
    }
    float s2 = b2[0];
#pragma unroll
    for (int c = 0; c < NHID; ++c) s2 += fmaxf(p[c] + b1[c], 0.f) * w2[c];
    return 1.f / (1.f + expf(-s2));
}

__global__ void k_node_mlps(const float* __restrict__ x,
                            const float* cw1, const float* cb1,
                            const float* cw2, const float* cb2,
                            const float* fw1, const float* fb1,
                            const float* fw2, const float* fb2,
                            float* __restrict__ kappa, float* __restrict__ fout)
{
    int lane = threadIdx.x & 31;
    int node = (blockIdx.x * blockDim.x + threadIdx.x) >> 5;
    if (node >= NN) return;
    int kb = lane * 4;
    float xr[4];
#pragma unroll
    for (int j = 0; j < 4; ++j) xr[j] = x[node * DD + kb + j];
    float kv = mlp_head(xr, kb, cw1, cb1, cw2, cb2);
    float fv = mlp_head(xr, kb, fw1, fb1, fw2, fb2);
    if (lane == 0) { kappa[node] = kv; fout[node] = fv; }
}

// ------------------------------------------------------- edge-weight MLP (big vec-mat)
__global__ void k_wm1(const int* __restrict__ edge_attr,
                      const float* __restrict__ wm_w1,
                      float* __restrict__ whid_raw)
{
    __shared__ float sdata[256];
    int tid = threadIdx.x;
    int c = tid & 63;
    int group = (blockIdx.x * blockDim.x + tid) >> 6;
    int ngroups = (gridDim.x * blockDim.x) >> 6;
    float acc = 0.f;
    for (int e = group; e < EE; e += ngroups) {
        float ea = (float)(edge_attr[e * 3 + 0] + edge_attr[e * 3 + 1] + edge_attr[e * 3 + 2]);
        acc += ea * wm_w1[(size_t)e * WHID + c];
    }
    sdata[tid] = acc;
    __syncthreads();
    if (tid < 64) {
        float t = sdata[tid] + sdata[tid + 64] + sdata[tid + 128] + sdata[tid + 192];
        unsafeAtomicAdd(&whid_raw[tid], t);
    }
}

__global__ void k_wm2(const float* __restrict__ whid_raw, const float* wm_b1,
                      const float* __restrict__ wm_w2, const float* wm_b2,
                      float* __restrict__ whid2)
{
    __shared__ float h1[WHID];
    int t = threadIdx.x;
    h1[t] = fmaxf(whid_raw[t] + wm_b1[t], 0.f);
    __syncthreads();
    float s = wm_b2[t];
#pragma unroll 8
    for (int k = 0; k < WHID; ++k) s += h1[k] * wm_w2[k * WHID + t];
    whid2[t] = fmaxf(s, 0.f);
}

__global__ void k_weights(const float* __restrict__ whid2,
                          const float* __restrict__ wm_w3,
                          const float* __restrict__ wm_b3,
                          float* __restrict__ wts)
{
    __shared__ float sw[WHID];
    if (threadIdx.x < WHID) sw[threadIdx.x] = whid2[threadIdx.x];
    __syncthreads();
    for (int e = blockIdx.x * blockDim.x + threadIdx.x; e < EE; e += gridDim.x * blockDim.x) {
        float s = wm_b3[e];
#pragma unroll 8
        for (int c = 0; c < WHID; ++c) s += sw[c] * wm_w3[(size_t)c * EE + e];
        wts[e] = 1.f / (1.f + expf(-s));
    }
}

// ------------------------------------------------------- curvature segment sums
__global__ void k_seg1(const int* __restrict__ ei, const float* __restrict__ f,
                       const float* __restrict__ w,
                       float* __restrict__ gamma, float* __restrict__ delta_f)
{
    int e = blockIdx.x * blockDim.x + threadIdx.x;
    if (e >= EE) return;
    int s = ei[e], d = ei[EE + e];
    float fd = f[d] - f[s];
    float wv = w[e];
    unsafeAtomicAdd(&gamma[s], 0.5f * wv * fd * fd);
    unsafeAtomicAdd(&delta_f[s], wv * fd);
}

__global__ void k_seg2(const int* __restrict__ ei, const float* __restrict__ f,
                       const float* __restrict__ w,
                       const float* __restrict__ gamma, const float* __restrict__ delta_f,
                       float* __restrict__ dg, float* __restrict__ gfd)
{
    int e = blockIdx.x * blockDim.x + threadIdx.x;
    if (e >= EE) return;
    int s = ei[e], d = ei[EE + e];
    float fd = f[d] - f[s];
    float wv = w[e];
    unsafeAtomicAdd(&dg[s], wv * (gamma[d] - gamma[s]));
    unsafeAtomicAdd(&gfd[s], 0.5f * wv * fd * (delta_f[d] - delta_f[s]));
}

__global__ void k_gamma2(const float* __restrict__ dg, const float* __restrict__ gfd,
                         float* __restrict__ gamma2)
{
    int n = blockIdx.x * blockDim.x + threadIdx.x;
    if (n >= NN) return;
    gamma2[n] = 0.5f * dg[n] - gfd[n];
}

// ------------------------------------------------------- O(N^2) curvature loss
__global__ void k_loss(const float* __restrict__ kappa, const float* __restrict__ gamma,
                       const float* __restrict__ gamma2, float* __restrict__ loss)
{
    __shared__ float sg[2048], sg2[2048];
    int i = blockIdx.x * blockDim.x + threadIdx.x;
    float ki = (i < NN) ? kappa[i] : 0.f;
    float acc = 0.f;
    for (int jt = 0; jt < NN; jt += 2048) {
        int cnt = min(2048, NN - jt);
        for (int j = threadIdx.x; j < cnt; j += blockDim.x) {
            sg[j] = gamma[jt + j]; sg2[j] = gamma2[jt + j];
        }
        __syncthreads();
        if (i < NN)
            for (int j = 0; j < cnt; ++j)
                acc += fmaxf(ki * sg[j] - sg2[j], 0.f);
        __syncthreads();
    }
    if (i < NN) unsafeAtomicAdd(loss, acc - ki);
}

// ------------------------------------------------------- top-100 mask (lax.top_k ties: lower idx)
__global__ void __launch_bounds__(1024) k_topk(const float* __restrict__ kappa,
                                               float* __restrict__ vm)
{
    __shared__ float skap[NN];
    __shared__ unsigned char sel[NN];
    __shared__ float rv[1024];
    __shared__ int   ri[1024];
    int t = threadIdx.x;
    for (int n = t; n < NN; n += 1024) { skap[n] = kappa[n]; sel[n] = 0; }
    __syncthreads();
    for (int it = 0; it < TOPK; ++it) {
        float bv = -3.0e38f; int bi = NN;
        for (int n = t; n < NN; n += 1024) {
            if (!sel[n]) {
                float v = skap[n];
                if (v > bv) { bv = v; bi = n; }   // first (lowest n) wins ties
            }
        }
        rv[t] = bv; ri[t] = bi;
        __syncthreads();
        for (int st = 512; st > 0; st >>= 1) {
            if (t < st) {
                float ov = rv[t + st]; int oi = ri[t + st];
                if (ov > rv[t] || (ov == rv[t] && oi < ri[t])) { rv[t] = ov; ri[t] = oi; }
            }
            __syncthreads();
        }
        if (t == 0) sel[ri[0]] = 1;
        __syncthreads();
    }
    for (int n = t; n < NN; n += 1024) vm[n] = sel[n] ? 0.f : 1.f;
}

__global__ void k_mew(const int* __restrict__ ei, const float* __restrict__ vm,
                      const float* __restrict__ new_w, float* __restrict__ cw)
{
    int e = blockIdx.x * blockDim.x + threadIdx.x;
    if (e >= EE) return;
    int s = ei[e], d = ei[EE + e];
    float m = (vm[s] == 0.f || vm[d] == 0.f) ? 1e-5f : 1.f;
    cw[e] = m * new_w[e];
}

// ------------------------------------------------------- message + scatter-add
__global__ void k_msg(const int* __restrict__ ei, const int* __restrict__ eattr,
                      const float* __restrict__ cw, const float* __restrict__ bond,
                      const float* __restrict__ h, float* __restrict__ aggr)
{
    int t = blockIdx.x * blockDim.x + threadIdx.x;
    int e = t >> 5, lane = t & 31;
    if (e >= EE) return;
    int s = ei[e], d = ei[EE + e];
    float c = cw[e];
    const int* ea = eattr + e * 3;
    const float* b0 = bond + ea[0] * DD;
    const float* b1 = bond + BONDV * DD + ea[1] * DD;
    const float* b2 = bond + 2 * BONDV * DD + ea[2] * DD;
    int fb = lane * 4;
    const float4 hv = *(const float4*)(h + (size_t)s * DD + fb);
    float v0 = fmaxf(hv.x + b0[fb + 0] + b1[fb + 0] + b2[fb + 0], 0.f);
    float v1 = fmaxf(hv.y + b0[fb + 1] + b1[fb + 1] + b2[fb + 1], 0.f);
    float v2 = fmaxf(hv.z + b0[fb + 2] + b1[fb + 2] + b2[fb + 2], 0.f);
    float v3 = fmaxf(hv.w + b0[fb + 3] + b1[fb + 3] + b2[fb + 3], 0.f);
    float* ap = aggr + (size_t)d * DD + fb;
    unsafeAtomicAdd(ap + 0, c * v0);
    unsafeAtomicAdd(ap + 1, c * v1);
    unsafeAtomicAdd(ap + 2, c * v2);
    unsafeAtomicAdd(ap + 3, c * v3);
}

// ------------------------------------------------------- f32 WMMA GEMM (16x16x4)
// mode 0: A[m][k] = eps*nsl[m]*A0[m][k] + A1[m][k]          (self-loop combine)
// mode 1: A[m][k] = relu(A0[m][k]*scale[k] + shift[k])       (bn + relu)
// Y = A @ W + bias.   One wave per 16x16 tile, 8 waves/block share a 16x128 A strip.
__global__ void __launch_bounds__(256) k_gemm(const float* __restrict__ A0,
                                              const float* __restrict__ A1,
                                              const float* __restrict__ p0,
                                              const float* __restrict__ p1,
                                              const float* __restrict__ W,
                                              const float* __restrict__ bias,
                                              float* __restrict__ Y, int mode)
{
    __shared__ float sA[16 * 132];          // padded stride -> bank-conflict free
    int m0 = blockIdx.x * 16;
    for (int idx = threadIdx.x; idx < 16 * DD; idx += blockDim.x) {
        int r = idx >> 7, k = idx & 127;
        int gr = m0 + r;
        float v;
        if (mode == 0)
            v = p1[0] * p0[gr] * A0[(size_t)gr * DD + k] + A1[(size_t)gr * DD + k];
        else
            v = fmaxf(A0[(size_t)gr * DD + k] * p0[k] + p1[k], 0.f);
        sA[r * 132 + k] = v;
    }
    __syncthreads();

    int lane = threadIdx.x & 31;
    int wv   = threadIdx.x >> 5;
    int col0 = wv * 16;
    int half = lane >> 4;          // K-pair selector (ISA 16x4 / 4x16 layout)
    int lr   = lane & 15;          // row (A) / col (B) within tile
    v8f c = {};
#pragma unroll 8
    for (int k = 0; k < DD; k += 4) {
        int kb = k + half * 2;
        v2f a, b;
        a.x = sA[lr * 132 + kb];
        a.y = sA[lr * 132 + kb + 1];
        b.x = W[(size_t)kb * DD + col0 + lr];
        b.y = W[(size_t)(kb + 1) * DD + col0 + lr];
        c = __builtin_amdgcn_wmma_f32_16x16x4_f32(false, a, false, b, (short)0, c,
                                                  false, false);
    }
    int col = col0 + lr;
    float bs = bias[col];
#pragma unroll
    for (int r = 0; r < 8; ++r)
        Y[(size_t)(m0 + r + half * 8) * DD + col] = c[r] + bs;
}

// ------------------------------------------------------- BN stats (block per column)
__global__ void k_bnstats(const float* __restrict__ Y, const float* __restrict__ g,
                          const float* __restrict__ b,
                          float* __restrict__ scale, float* __restrict__ shift)
{
    __shared__ float ss[256], sq[256];
    int c = blockIdx.x;
    float s = 0.f, s2 = 0.f;
    for (int n = threadIdx.x; n < NN; n += blockDim.x) {
        float v = Y[(size_t)n * DD + c];
        s += v; s2 += v * v;
    }
    ss[threadIdx.x] = s; sq[threadIdx.x] = s2;
    __syncthreads();
    for (int st = 128; st > 0; st >>= 1) {
        if (threadIdx.x < st) {
            ss[threadIdx.x] += ss[threadIdx.x + st];
            sq[threadIdx.x] += sq[threadIdx.x + st];
        }
        __syncthreads();
    }
    if (threadIdx.x == 0) {
        float mean = ss[0] / (float)NN;
        float var  = sq[0] / (float)NN - mean * mean;
        float rs   = rsqrtf(var + 1e-5f);
        float sc   = g[c] * rs;
        scale[c] = sc;
        shift[c] = b[c] - mean * sc;
    }
}

// ------------------------------------------------------- final BN (+relu) + accumulate
__global__ void k_bn_acc(const float* __restrict__ Z, const float* __restrict__ scale,
                         const float* __restrict__ shift,
                         float* __restrict__ hout, float* __restrict__ nrep, int dorelu)
{
    int i = blockIdx.x * blockDim.x + threadIdx.x;
    if (i >= NN * DD) return;
    int d = i & 127;
    float v = Z[i] * scale[d] + shift[d];
    if (dorelu) v = fmaxf(v, 0.f);
    hout[i] = v;
    nrep[i] += v;
}

// ================================================================ host
extern "C" void kernel_launch(void* const* d_in, const int* in_sizes, int n_in,
                              void* d_out, int out_size, void* d_ws, size_t ws_size,
                              hipStream_t stream)
{
    (void)in_sizes; (void)n_in; (void)out_size; (void)ws_size;
    const int*   x_feat = (const int*)d_in[0];
    const int*   ei     = (const int*)d_in[1];
    const int*   eattr  = (const int*)d_in[2];
    const float* n_ew   = (const float*)d_in[3];
    const float* n_sl   = (const float*)d_in[4];
    const float* atom   = (const float*)d_in[5];
    const float* cw1p   = (const float*)d_in[6];
    const float* cb1p   = (const float*)d_in[7];
    const float* cw2p   = (const float*)d_in[8];
    const float* cb2p   = (const float*)d_in[9];
    const float* fw1p   = (const float*)d_in[10];
    const float* fb1p   = (const float*)d_in[11];
    const float* fw2p   = (const float*)d_in[12];
    const float* fb2p   = (const float*)d_in[13];
    const float* wmw1   = (const float*)d_in[14];
    const float* wmb1   = (const float*)d_in[15];
    const float* wmw2   = (const float*)d_in[16];
    const float* wmb2   = (const float*)d_in[17];
    const float* wmw3   = (const float*)d_in[18];
    const float* wmb3   = (const float*)d_in[19];
    const float* bond   = (const float*)d_in[20];
    const float* convw1 = (const float*)d_in[21];
    const float* convb1 = (const float*)d_in[22];
    const float* cbn_g  = (const float*)d_in[23];
    const float* cbn_b  = (const float*)d_in[24];
    const float* convw2 = (const float*)d_in[25];
    const float* convb2 = (const float*)d_in[26];
    const float* c_eps  = (const float*)d_in[27];
    const float* bn_g   = (const float*)d_in[28];
    const float* bn_b   = (const float*)d_in[29];

    float* out = (float*)d_out;
    float* ws  = (float*)d_ws;
    size_t off = 0;
    auto alloc = [&](size_t n) { float* p = ws + off; off += n; return p; };
    float* bufA = alloc((size_t)NN * DD);
    float* bufB = alloc((size_t)NN * DD);
    float* aggr = alloc((size_t)NN * DD);   // also reused as Z (gemm2 output)
    float* Y1   = alloc((size_t)NN * DD);
    float* wts  = alloc(EE);
    float* cwe  = alloc(EE);
    float* kap  = alloc(NN);
    float* fno  = alloc(NN);
    float* gam  = alloc(NN);                // gam, dfb, dgb, gfd contiguous (one memset)
    float* dfb  = alloc(NN);
    float* dgb  = alloc(NN);
    float* gfd  = alloc(NN);
    float* gam2 = alloc(NN);
    float* vm   = alloc(NN);
    float* whid_raw = alloc(WHID);
    float* whid2    = alloc(WHID);
    float* scale = alloc(DD);
    float* shift = alloc(DD);
    (void)dfb; (void)dgb; (void)gfd;

    hipMemsetAsync(whid_raw, 0, WHID * sizeof(float), stream);
    hipMemsetAsync(gam, 0, 4 * NN * sizeof(float), stream);
    hipMemsetAsync(out + (size_t)NN * DD, 0, sizeof(float), stream);

    k_embed<<<(NN * DD + 255) / 256, 256, 0, stream>>>(x_feat, atom, bufA, out);
    k_node_mlps<<<(NN * 32 + 255) / 256, 256, 0, stream>>>(
        bufA, cw1p, cb1p, cw2p, cb2p, fw1p, fb1p, fw2p, fb2p, kap, fno);
    k_wm1<<<1024, 256, 0, stream>>>(eattr, wmw1, whid_raw);
    k_wm2<<<1, WHID, 0, stream>>>(whid_raw, wmb1, wmw2, wmb2, whid2);
    k_weights<<<2048, 256, 0, stream>>>(whid2, wmw3, wmb3, wts);
    k_seg1<<<(EE + 255) / 256, 256, 0, stream>>>(ei, fno, wts, gam, dfb);
    k_seg2<<<(EE + 255) / 256, 256, 0, stream>>>(ei, fno, wts, gam, dfb, dgb, gfd);
    k_gamma2<<<(NN + 255) / 256, 256, 0, stream>>>(dgb, gfd, gam2);
    k_loss<<<(NN + 255) / 256, 256, 0, stream>>>(kap, gam, gam2, out + (size_t)NN * DD);
    k_topk<<<1, 1024, 0, stream>>>(kap, vm);
    k_mew<<<(EE + 255) / 256, 256, 0, stream>>>(ei, vm, n_ew, cwe);

    float* hprev = bufA;
    float* hnew  = bufB;
    for (int l = 0; l < LL; ++l) {
        const float* cw = (l == 0) ? n_ew : cwe;     // k = 0 for l==0, else top-100 mask
        hipMemsetAsync(aggr, 0, (size_t)NN * DD * sizeof(float), stream);
        k_msg<<<(EE * 32) / 256, 256, 0, stream>>>(
            ei, eattr, cw, bond + (size_t)l * 3 * BONDV * DD, hprev, aggr);
        k_gemm<<<NN / 16, 256, 0, stream>>>(
            hprev, aggr, n_sl, c_eps + l, convw1 + (size_t)l * DD * DD,
            convb1 + l * DD, Y1, 0);
        k_bnstats<<<DD, 256, 0, stream>>>(Y1, cbn_g + l * DD, cbn_b + l * DD, scale, shift);
        k_gemm<<<NN / 16, 256, 0, stream>>>(
            Y1, nullptr, scale, shift, convw2 + (size_t)l * DD * DD,
            convb2 + l * DD, aggr, 1);
        k_bnstats<<<DD, 256, 0, stream>>>(aggr, bn_g + l * DD, bn_b + l * DD, scale, shift);
        k_bn_acc<<<(NN * DD + 255) / 256, 256, 0, stream>>>(
            aggr, scale, shift, hnew, out, (l < LL - 1) ? 1 : 0);
        float* tmp = hprev; hprev = hnew; hnew = tmp;
    }
}